// Isosplit6Clustering_49237505081531
// MI455X (gfx1250) — compile-verified
//
#include <hip/hip_runtime.h>
#include <hip/hip_bf16.h>

// MI455X (gfx1250) implementation. wave32, WMMA f32 16x16x4 for the
// covariance GEMM. All reductions are fixed-order (deterministic).

typedef __attribute__((ext_vector_type(2))) float v2f;
typedef __attribute__((ext_vector_type(8))) float v8f;

#define G1_BLOCKS 128   // partial-XtX blocks
#define WPB 8           // waves per block in k_xtx

// ---------- helpers ----------
__device__ __forceinline__ unsigned fkey(float f) {
    unsigned b = __float_as_uint(f);
    return (b & 0x80000000u) ? ~b : (b | 0x80000000u);   // order-preserving map
}
__device__ __forceinline__ float funkey(unsigned k) {
    unsigned b = (k & 0x80000000u) ? (k ^ 0x80000000u) : ~k;
    return __uint_as_float(b);
}

__global__ void k_zero(unsigned* p, int n) {
    for (int i = blockIdx.x * blockDim.x + threadIdx.x; i < n; i += gridDim.x * blockDim.x)
        p[i] = 0u;
}

// ---------- 1) XtX + column sums via f32 WMMA ----------
// A[m][k] = X[row k][dim m], B[k][n] = X[row k][dim n]  ->  identical VGPR layout:
// lane l, vgpr j  holds  X[row q*4 + (l>>4)*2 + j][dimoff + (l&15)]
__global__ __launch_bounds__(256) void k_xtx(const float* __restrict__ X, int nquads,
                                             float* __restrict__ xtxp, float* __restrict__ sump) {
    __shared__ float lds_c[WPB][1024];
    __shared__ float lds_s[WPB][32];

    const int lane  = threadIdx.x & 31;
    const int wave  = threadIdx.x >> 5;
    const int gwave = blockIdx.x * WPB + wave;
    const int nwav  = gridDim.x * WPB;
    const int rsel  = lane >> 4;     // 0: rows {0,1}, 1: rows {2,3} of the quad
    const int c0    = lane & 15;

    v8f c00 = {}, c01 = {}, c10 = {}, c11 = {};
    float s0 = 0.f, s1 = 0.f;

    for (int q = gwave; q < nquads; q += nwav) {
        const float* p = X + (size_t)q * 128 + rsel * 64;   // row q*4 + rsel*2
        __builtin_prefetch(p + (size_t)nwav * 128, 0, 0);   // global_prefetch_b8
        v2f a0, a1;
        a0.x = p[c0];        a0.y = p[32 + c0];        // dims 0..15, rows +0,+1
        a1.x = p[16 + c0];   a1.y = p[48 + c0];        // dims 16..31
        c00 = __builtin_amdgcn_wmma_f32_16x16x4_f32(false, a0, false, a0, (short)0, c00, false, false);
        c01 = __builtin_amdgcn_wmma_f32_16x16x4_f32(false, a0, false, a1, (short)0, c01, false, false);
        c10 = __builtin_amdgcn_wmma_f32_16x16x4_f32(false, a1, false, a0, (short)0, c10, false, false);
        c11 = __builtin_amdgcn_wmma_f32_16x16x4_f32(false, a1, false, a1, (short)0, c11, false, false);
        s0 += a0.x + a0.y;
        s1 += a1.x + a1.y;
    }

    // combine the two half-wave row groups (same columns)
    s0 += __shfl_xor(s0, 16);
    s1 += __shfl_xor(s1, 16);
    if (lane < 16) { lds_s[wave][c0] = s0; lds_s[wave][16 + c0] = s1; }

    // C layout: lanes 0-15: M = v, lanes 16-31: M = 8+v; N = lane&15
    float* cw = lds_c[wave];
#pragma unroll
    for (int v = 0; v < 8; v++) {
        int m = v + 8 * rsel, n = c0;
        cw[(m)      * 32 + (n)]      = c00[v];
        cw[(m)      * 32 + (16 + n)] = c01[v];
        cw[(16 + m) * 32 + (n)]      = c10[v];
        cw[(16 + m) * 32 + (16 + n)] = c11[v];
    }
    __syncthreads();

    for (int idx = threadIdx.x; idx < 1024; idx += 256) {
        float a = 0.f;
#pragma unroll
        for (int w = 0; w < WPB; w++) a += lds_c[w][idx];
        xtxp[(size_t)blockIdx.x * 1024 + idx] = a;
    }
    if (threadIdx.x < 32) {
        float a = 0.f;
#pragma unroll
        for (int w = 0; w < WPB; w++) a += lds_s[w][threadIdx.x];
        sump[blockIdx.x * 32 + threadIdx.x] = a;
    }
}

// ---------- 2) reduce partials -> covariance ----------
__global__ __launch_bounds__(1024) void k_cov(const float* __restrict__ X,
                                              const float* __restrict__ xtxp,
                                              const float* __restrict__ sump,
                                              int nrows, float* cov, float* meanOut) {
    __shared__ float mu[32];
    const int t = threadIdx.x;
    const int nq4 = (nrows / 4) * 4;
    if (t < 32) {
        float s = 0.f;
        for (int b = 0; b < G1_BLOCKS; b++) s += sump[b * 32 + t];
        for (int r = nq4; r < nrows; r++) s += X[(size_t)r * 32 + t];   // tail
        float m = s / (float)nrows;
        mu[t] = m; meanOut[t] = m;
    }
    __syncthreads();
    float s = 0.f;
    for (int b = 0; b < G1_BLOCKS; b++) s += xtxp[(size_t)b * 1024 + t];
    const int i = t >> 5, j = t & 31;
    for (int r = nq4; r < nrows; r++) s += X[(size_t)r * 32 + i] * X[(size_t)r * 32 + j];
    cov[t] = (s - (float)nrows * mu[i] * mu[j]) / (float)(nrows - 1);
}

// ---------- 3) 32x32 Jacobi eigendecomposition (single wave) ----------
__global__ __launch_bounds__(32) void k_eigh(const float* cov, const float* meanp,
                                             float* Vtop, float* projmu) {
    __shared__ float A[32][32], V[32][32], ev[32];
    __shared__ int idx[10];
    const int i = threadIdx.x;
    for (int j = 0; j < 32; j++) { A[i][j] = cov[i * 32 + j]; V[i][j] = (i == j) ? 1.f : 0.f; }
    __syncthreads();

    for (int sweep = 0; sweep < 14; sweep++) {
        for (int p = 0; p < 31; p++) {
            for (int q = p + 1; q < 32; q++) {
                float app = A[p][p], aqq = A[q][q], apq = A[p][q];
                float c, s;
                if (fabsf(apq) > 1e-22f) {
                    float tau = (aqq - app) / (2.f * apq);
                    float tt = ((tau >= 0.f) ? 1.f : -1.f) / (fabsf(tau) + sqrtf(1.f + tau * tau));
                    c = rsqrtf(1.f + tt * tt); s = tt * c;
                } else { c = 1.f; s = 0.f; }
                float aip = A[i][p], aiq = A[i][q];
                __syncthreads();
                A[i][p] = c * aip - s * aiq;            // A <- A*J (columns)
                A[i][q] = s * aip + c * aiq;
                float vip = V[i][p], viq = V[i][q];
                V[i][p] = c * vip - s * viq;            // V <- V*J
                V[i][q] = s * vip + c * viq;
                __syncthreads();
                float apj = A[p][i], aqj = A[q][i];
                __syncthreads();
                A[p][i] = c * apj - s * aqj;            // A <- J^T*A (rows)
                A[q][i] = s * apj + c * aqj;
                __syncthreads();
            }
        }
    }
    ev[i] = A[i][i];
    __syncthreads();
    if (i == 0) {                                        // top-10, descending
        unsigned used = 0;
        for (int c = 0; c < 10; c++) {
            int best = 0; float bv = -3.0e38f;
            for (int j = 0; j < 32; j++)
                if (!((used >> j) & 1u) && ev[j] > bv) { bv = ev[j]; best = j; }
            idx[c] = best; used |= (1u << best);
        }
    }
    __syncthreads();
    for (int c = 0; c < 10; c++) Vtop[i * 10 + c] = V[i][idx[c]];
    if (i < 10) {
        float s = 0.f;
        for (int d = 0; d < 32; d++) s += meanp[d] * V[d][idx[i]];
        projmu[i] = s;                                   // mean projected: feats = X*V - mu*V
    }
}

// ---------- 4) projection: feats = X*Vtop - projmu ----------
__global__ __launch_bounds__(256) void k_proj(const float* __restrict__ X,
                                              const float* __restrict__ Vtop,
                                              const float* __restrict__ projmu,
                                              float* __restrict__ out, int nrows) {
    __shared__ float Vs[320], pm[10];
    for (int t = threadIdx.x; t < 320; t += blockDim.x) Vs[t] = Vtop[t];
    for (int t = threadIdx.x; t < 10; t += blockDim.x) pm[t] = projmu[t];
    __syncthreads();
    for (int r = blockIdx.x * blockDim.x + threadIdx.x; r < nrows; r += gridDim.x * blockDim.x) {
        const float4* row = (const float4*)(X + (size_t)r * 32);
        float acc[10];
#pragma unroll
        for (int c = 0; c < 10; c++) acc[c] = -pm[c];
#pragma unroll
        for (int d4 = 0; d4 < 8; d4++) {
            float4 x = row[d4];
            const float* v = &Vs[d4 * 40];
#pragma unroll
            for (int c = 0; c < 10; c++)
                acc[c] += x.x * v[c] + x.y * v[10 + c] + x.z * v[20 + c] + x.w * v[30 + c];
        }
        float* o = out + (size_t)r * 10;
#pragma unroll
        for (int c = 0; c < 10; c++) o[c] = acc[c];
    }
}

// ---------- 5) medians: 4-pass radix selection ----------
__global__ void k_count(const int* __restrict__ labels, unsigned* cnt, int nrows) {
    for (int r = blockIdx.x * blockDim.x + threadIdx.x; r < nrows; r += gridDim.x * blockDim.x)
        atomicAdd(&cnt[labels[r] - 1], 1u);
}

__global__ __launch_bounds__(512) void k_selinit(const unsigned* cnt, unsigned* rank, unsigned* targ) {
    const int i = threadIdx.x;            // 512 = 16 clusters * 32 dims
    unsigned n = cnt[i >> 5];
    rank[i] = n ? (n - 1u) >> 1 : 0u;     // lower median rank
    targ[i] = 0u;
}

// grid: (chunks, 16 dim-groups of 2 dims); LDS hist = 16*2*256*4 = 32 KB
__global__ __launch_bounds__(256) void k_hist(const float* __restrict__ X,
                                              const int* __restrict__ labels,
                                              const unsigned* __restrict__ targ,
                                              unsigned* __restrict__ hist,
                                              int pass, int nrows) {
    __shared__ unsigned h[16 * 2 * 256];
    __shared__ unsigned tg[16][2];
    const int dg = blockIdx.y;
    for (int i = threadIdx.x; i < 16 * 2 * 256; i += 256) h[i] = 0u;
    for (int i = threadIdx.x; i < 32; i += 256)
        tg[i >> 1][i & 1] = targ[(i >> 1) * 32 + dg * 2 + (i & 1)];
    __syncthreads();

    const int shiftKnown = 32 - 8 * pass;
    const int shiftBin   = 24 - 8 * pass;
    for (int r = blockIdx.x * 256 + threadIdx.x; r < nrows; r += gridDim.x * 256) {
        int k = labels[r] - 1;
#pragma unroll
        for (int j = 0; j < 2; j++) {
            unsigned key = fkey(X[(size_t)r * 32 + dg * 2 + j]);
            bool ok = (pass == 0) || ((key >> shiftKnown) == tg[k][j]);
            if (ok) atomicAdd(&h[((k * 2 + j) << 8) | ((key >> shiftBin) & 255u)], 1u);
        }
    }
    __syncthreads();
    for (int i = threadIdx.x; i < 16 * 2 * 256; i += 256) {
        unsigned v = h[i];
        if (v) {
            int k = i >> 9, j = (i >> 8) & 1, bin = i & 255;
            atomicAdd(&hist[(((k * 32) + dg * 2 + j) << 8) | bin], v);
        }
    }
}

__global__ __launch_bounds__(512) void k_select(const unsigned* __restrict__ hist,
                                                unsigned* rank, unsigned* targ) {
    const int i = threadIdx.x;
    unsigned r = rank[i];
    const unsigned* hb = hist + ((size_t)i << 8);
    unsigned cum = 0, bin = 255;
    for (int b = 0; b < 256; b++) {
        unsigned c = hb[b];
        if (r < cum + c) { bin = (unsigned)b; break; }
        cum += c;
    }
    targ[i] = (targ[i] << 8) | bin;
    rank[i] = r - cum;
}

__global__ __launch_bounds__(512) void k_medfin(const unsigned* targ, float* med) {
    med[threadIdx.x] = funkey(targ[threadIdx.x]);
}

// ---------- 6) cdist + Prim MST + cut (faithful to reference) ----------
__global__ __launch_bounds__(256) void k_mst(const float* __restrict__ med, int* grp) {
    __shared__ float M[16][32];
    __shared__ float D[16][16];
    for (int i = threadIdx.x; i < 512; i += 256) M[i >> 5][i & 31] = med[i];
    __syncthreads();
    {
        int i = threadIdx.x >> 4, j = threadIdx.x & 15;
        float s = 0.f;
        for (int d = 0; d < 32; d++) { float df = M[i][d] - M[j][d]; s += df * df; }
        D[i][j] = sqrtf(s);
    }
    __syncthreads();
    if (threadIdx.x == 0) {
        bool in[16]; float me[16]; int par[16];
        int us[15], vs[15]; float we[15];
        for (int j = 0; j < 16; j++) { in[j] = false; me[j] = D[0][j]; par[j] = 0; }
        in[0] = true; me[0] = __builtin_inff();
        for (int st = 0; st < 15; st++) {
            int v = 0; float bv = __builtin_inff();
            for (int j = 0; j < 16; j++) {
                float t = in[j] ? __builtin_inff() : me[j];
                if (t < bv) { bv = t; v = j; }            // first-min == jnp.argmin
            }
            us[st] = par[v]; vs[st] = v; we[st] = me[v]; in[v] = true;
            for (int j = 0; j < 16; j++) {
                float nd = D[v][j];
                if (nd < me[j]) { me[j] = nd; par[j] = v; }
            }
        }
        int longest = 0; float bw = we[0];
        for (int e = 1; e < 15; e++) if (we[e] > bw) { bw = we[e]; longest = e; }
        bool adj[16][16];
        for (int a = 0; a < 16; a++) for (int b = 0; b < 16; b++) adj[a][b] = false;
        for (int e = 0; e < 15; e++)
            if (e != longest) { adj[us[e]][vs[e]] = true; adj[vs[e]][us[e]] = true; }
        bool reach[16];
        for (int a = 0; a < 16; a++) reach[a] = false;
        reach[us[longest]] = true;
        for (int it = 0; it < 16; it++) {
            bool nr[16];
            for (int a = 0; a < 16; a++) {
                bool any = reach[a];
                for (int b = 0; b < 16; b++) any = any || (adj[a][b] && reach[b]);
                nr[a] = any;
            }
            for (int a = 0; a < 16; a++) reach[a] = nr[a];
        }
        for (int k = 0; k < 16; k++) grp[k] = reach[k] ? 1 : 0;
    }
}

// ---------- 7) final two-way partition ----------
__global__ void k_part(const int* __restrict__ labels, const int* __restrict__ grp,
                       int* __restrict__ part, int nrows) {
    for (int r = blockIdx.x * blockDim.x + threadIdx.x; r < nrows; r += gridDim.x * blockDim.x)
        part[r] = grp[labels[r] - 1] ? 1 : 2;
}

// ---------- launch ----------
extern "C" void kernel_launch(void* const* d_in, const int* in_sizes, int n_in,
                              void* d_out, int out_size, void* d_ws, size_t ws_size,
                              hipStream_t stream) {
    (void)n_in; (void)out_size; (void)ws_size;
    const float* X      = (const float*)d_in[0];   // [N,32]
    const int*   labels = (const int*)d_in[1];     // [N], values 1..16
    const int    N      = in_sizes[1];

    int*   part  = (int*)d_out;                    // output 0: [N] int32
    float* feats = (float*)d_out + N;              // output 1: [N,10] f32

    // workspace layout (floats then uints), ~1.1 MB total
    float* wf    = (float*)d_ws;
    float* xtxp  = wf;                               // G1*1024
    float* sump  = xtxp + (size_t)G1_BLOCKS * 1024;  // G1*32
    float* cov   = sump + (size_t)G1_BLOCKS * 32;    // 1024
    float* meanp = cov + 1024;                       // 32
    float* Vtop  = meanp + 32;                       // 320
    float* pmu   = Vtop + 320;                       // 16
    float* med   = pmu + 16;                         // 512
    unsigned* cnt  = (unsigned*)(med + 512);         // 16
    unsigned* rank = cnt + 16;                       // 512
    unsigned* targ = rank + 512;                     // 512
    int*      grp  = (int*)(targ + 512);             // 16
    unsigned* hist = (unsigned*)(grp + 16);          // 512*256

    k_zero<<<1, 64, 0, stream>>>(cnt, 16);
    k_xtx<<<G1_BLOCKS, 256, 0, stream>>>(X, N / 4, xtxp, sump);
    k_cov<<<1, 1024, 0, stream>>>(X, xtxp, sump, N, cov, meanp);
    k_eigh<<<1, 32, 0, stream>>>(cov, meanp, Vtop, pmu);
    k_proj<<<512, 256, 0, stream>>>(X, Vtop, pmu, feats, N);

    k_count<<<256, 256, 0, stream>>>(labels, cnt, N);
    k_selinit<<<1, 512, 0, stream>>>(cnt, rank, targ);
    for (int p = 0; p < 4; p++) {
        k_zero<<<512, 256, 0, stream>>>(hist, 512 * 256);
        dim3 g(64, 16);
        k_hist<<<g, 256, 0, stream>>>(X, labels, targ, hist, p, N);
        k_select<<<1, 512, 0, stream>>>(hist, rank, targ);
    }
    k_medfin<<<1, 512, 0, stream>>>(targ, med);
    k_mst<<<1, 256, 0, stream>>>(med, grp);
    k_part<<<512, 256, 0, stream>>>(labels, grp, part, N);
}